// Model_35253091565755
// MI455X (gfx1250) — compile-verified
//
#include <hip/hip_runtime.h>
#include <hip/hip_bf16.h>

typedef float v2f __attribute__((ext_vector_type(2)));
typedef float v8f __attribute__((ext_vector_type(8)));

#define N_NODES 50000
#define N_EDGES 600000
#define FEAT    128

// ---------------------------------------------------------------------------
// Scatter: for each edge e, nsum[dst[e]] += X[src[e]] (128 floats), deg[dst]++.
// One wave (32 lanes) per edge, float4 per lane, HW fp32 atomics.
// ---------------------------------------------------------------------------
__global__ __launch_bounds__(256) void scatter_kernel(
    const float* __restrict__ X, const int* __restrict__ src,
    const int* __restrict__ dst, float* __restrict__ nsum,
    float* __restrict__ deg /* may be null */, int n_edges)
{
    const int wave = threadIdx.x >> 5;
    const int lane = threadIdx.x & 31;
    const int e = blockIdx.x * 8 + wave;
    if (e >= n_edges) return;
    const int s = src[e];
    const int d = dst[e];
    const float4 v =
        reinterpret_cast<const float4*>(X + (size_t)s * FEAT)[lane];
    float* base = nsum + (size_t)d * FEAT + lane * 4;
    unsafeAtomicAdd(base + 0, v.x);
    unsafeAtomicAdd(base + 1, v.y);
    unsafeAtomicAdd(base + 2, v.z);
    unsafeAtomicAdd(base + 3, v.w);
    if (deg != nullptr && lane == 0) unsafeAtomicAdd(deg + d, 1.0f);
}

// ---------------------------------------------------------------------------
// hnorm = (nsum + x) / (deg + 1)      [gcn aggregator normalization]
// ---------------------------------------------------------------------------
__global__ __launch_bounds__(256) void norm_gcn_kernel(
    const float* __restrict__ nsum, const float* __restrict__ X,
    const float* __restrict__ deg, float* __restrict__ hnorm)
{
    const int i = blockIdx.x * 256 + threadIdx.x;   // over N_NODES*FEAT
    const int node = i >> 7;
    const float dinv = 1.0f / (deg[node] + 1.0f);
    hnorm[i] = (nsum[i] + X[i]) * dinv;
}

// ---------------------------------------------------------------------------
// mean = nsum / max(deg, 1)           [mean aggregator normalization]
// ---------------------------------------------------------------------------
__global__ __launch_bounds__(256) void norm_mean_kernel(
    const float* __restrict__ nsum, const float* __restrict__ deg,
    float* __restrict__ mean)
{
    const int i = blockIdx.x * 256 + threadIdx.x;
    const int node = i >> 7;
    const float dinv = 1.0f / fmaxf(deg[node], 1.0f);
    mean[i] = nsum[i] * dinv;
}

// ---------------------------------------------------------------------------
// GEMM1: H1 = relu(A @ W + b)   A:[N_NODES,128] W:[128,128]
// One wave per 16x16 output tile; 8 waves/block cover the full N=128.
// v_wmma_f32_16x16x4_f32, fp32 end to end.
// A frag (16x4 f32): lane l -> row l&15, K pair kb=(l>>4)*2
// B frag (4x16 f32): lane l -> col l&15, same K pair
// C/D: vgpr r -> row r + 8*(l>=16), col l&15
// ---------------------------------------------------------------------------
__global__ __launch_bounds__(256) void gemm1_kernel(
    const float* __restrict__ A, const float* __restrict__ W,
    const float* __restrict__ bias, float* __restrict__ H1)
{
    const int tm   = blockIdx.x;           // 0..3124
    const int tn   = threadIdx.x >> 5;     // 0..7 (wave id)
    const int lane = threadIdx.x & 31;
    const int mrow = tm * 16 + (lane & 15);
    const int n    = tn * 16 + (lane & 15);
    const int kb   = (lane >> 4) * 2;      // 0 or 2

    const float* arow = A + (size_t)mrow * FEAT;
    v8f c = {};
    #pragma unroll 4
    for (int k0 = 0; k0 < FEAT; k0 += 4) {
        v2f a, b;
        a.x = arow[k0 + kb];
        a.y = arow[k0 + kb + 1];
        b.x = W[(size_t)(k0 + kb) * FEAT + n];
        b.y = W[(size_t)(k0 + kb + 1) * FEAT + n];
        c = __builtin_amdgcn_wmma_f32_16x16x4_f32(
                false, a, false, b, (short)0, c, false, false);
    }
    const float bn = bias[n];
    const int rbase = tm * 16 + ((lane >> 4) << 3);  // +8 for upper half-wave
    #pragma unroll
    for (int r = 0; r < 8; ++r) {
        float v = c[r] + bn;
        v = fmaxf(v, 0.0f);                          // relu
        H1[(size_t)(rbase + r) * FEAT + n] = v;
    }
}

// ---------------------------------------------------------------------------
// GEMM2: H2 = H1 @ Ws + MEAN @ Wn + b   (dual WMMA accumulate per K step)
// ---------------------------------------------------------------------------
__global__ __launch_bounds__(256) void gemm2_kernel(
    const float* __restrict__ H1, const float* __restrict__ MEAN,
    const float* __restrict__ Ws, const float* __restrict__ Wn,
    const float* __restrict__ bias, float* __restrict__ H2)
{
    const int tm   = blockIdx.x;
    const int tn   = threadIdx.x >> 5;
    const int lane = threadIdx.x & 31;
    const int mrow = tm * 16 + (lane & 15);
    const int n    = tn * 16 + (lane & 15);
    const int kb   = (lane >> 4) * 2;

    const float* arow = H1   + (size_t)mrow * FEAT;
    const float* mrow_p = MEAN + (size_t)mrow * FEAT;
    v8f c = {};
    #pragma unroll 4
    for (int k0 = 0; k0 < FEAT; k0 += 4) {
        v2f a, b;
        a.x = arow[k0 + kb];
        a.y = arow[k0 + kb + 1];
        b.x = Ws[(size_t)(k0 + kb) * FEAT + n];
        b.y = Ws[(size_t)(k0 + kb + 1) * FEAT + n];
        c = __builtin_amdgcn_wmma_f32_16x16x4_f32(
                false, a, false, b, (short)0, c, false, false);
        v2f a2, b2;
        a2.x = mrow_p[k0 + kb];
        a2.y = mrow_p[k0 + kb + 1];
        b2.x = Wn[(size_t)(k0 + kb) * FEAT + n];
        b2.y = Wn[(size_t)(k0 + kb + 1) * FEAT + n];
        c = __builtin_amdgcn_wmma_f32_16x16x4_f32(
                false, a2, false, b2, (short)0, c, false, false);
    }
    const float bn = bias[n];
    const int rbase = tm * 16 + ((lane >> 4) << 3);
    #pragma unroll
    for (int r = 0; r < 8; ++r)
        H2[(size_t)(rbase + r) * FEAT + n] = c[r] + bn;
}

// ---------------------------------------------------------------------------
// Edge scoring: score[e] = h2[src]@Wp_u + h2[dst]@Wp_v + bp   (OUT=2)
// One wave per edge; lane covers 4 features; wave32 shfl_xor reduction.
// ---------------------------------------------------------------------------
__global__ __launch_bounds__(256) void score_kernel(
    const float* __restrict__ H2, const float* __restrict__ Wpu,
    const float* __restrict__ Wpv, const float* __restrict__ bp,
    const int* __restrict__ src, const int* __restrict__ dst,
    float* __restrict__ out, int n_edges)
{
    const int wave = threadIdx.x >> 5;
    const int lane = threadIdx.x & 31;
    const int e = blockIdx.x * 8 + wave;
    if (e >= n_edges) return;
    const float* hs = H2 + (size_t)src[e] * FEAT;
    const float* hd = H2 + (size_t)dst[e] * FEAT;
    const int f0 = lane * 4;
    float s0 = 0.f, s1 = 0.f;
    #pragma unroll
    for (int j = 0; j < 4; ++j) {
        const int f = f0 + j;
        const float a = hs[f];
        const float b = hd[f];
        s0 += a * Wpu[f * 2 + 0] + b * Wpv[f * 2 + 0];
        s1 += a * Wpu[f * 2 + 1] + b * Wpv[f * 2 + 1];
    }
    #pragma unroll
    for (int off = 16; off > 0; off >>= 1) {
        s0 += __shfl_xor(s0, off, 32);
        s1 += __shfl_xor(s1, off, 32);
    }
    if (lane == 0) {
        out[(size_t)e * 2 + 0] = s0 + bp[0];
        out[(size_t)e * 2 + 1] = s1 + bp[1];
    }
}

// ---------------------------------------------------------------------------
extern "C" void kernel_launch(void* const* d_in, const int* in_sizes, int n_in,
                              void* d_out, int out_size, void* d_ws, size_t ws_size,
                              hipStream_t stream)
{
    const float* x   = (const float*)d_in[0];
    const float* W1  = (const float*)d_in[1];
    const float* b1  = (const float*)d_in[2];
    const float* W2s = (const float*)d_in[3];
    const float* W2n = (const float*)d_in[4];
    const float* b2  = (const float*)d_in[5];
    const float* Wpu = (const float*)d_in[6];
    const float* Wpv = (const float*)d_in[7];
    const float* bp  = (const float*)d_in[8];
    const int*   src = (const int*)d_in[9];
    const int*   dst = (const int*)d_in[10];
    float* out = (float*)d_out;

    // workspace layout (all 16B aligned)
    const size_t FEAT_BYTES = (size_t)N_NODES * FEAT * sizeof(float); // 25.6 MB
    char* ws = (char*)d_ws;
    float* nsum  = (float*)(ws);                              // 25.6 MB
    float* deg   = (float*)(ws + FEAT_BYTES);                 // 0.2 MB
    float* hnorm = (float*)(ws + FEAT_BYTES + 204800);        // 25.6 MB (-> mean)
    float* h1    = (float*)(ws + 2 * FEAT_BYTES + 204800);    // 25.6 MB
    float* h2    = nsum;  // reuse: nsum fully consumed before GEMM2 writes h2

    const int elem_blocks = (N_NODES * FEAT) / 256;   // 25000
    const int edge_blocks = N_EDGES / 8;              // 75000
    const int gemm_blocks = N_NODES / 16;             // 3125

    // ---- layer 1: gcn aggregate -> normalize -> GEMM+ReLU ----
    hipMemsetAsync(nsum, 0, FEAT_BYTES, stream);
    hipMemsetAsync(deg, 0, (size_t)N_NODES * sizeof(float), stream);
    scatter_kernel<<<edge_blocks, 256, 0, stream>>>(x, src, dst, nsum, deg, N_EDGES);
    norm_gcn_kernel<<<elem_blocks, 256, 0, stream>>>(nsum, x, deg, hnorm);
    gemm1_kernel<<<gemm_blocks, 256, 0, stream>>>(hnorm, W1, b1, h1);

    // ---- layer 2: mean aggregate -> normalize -> dual GEMM ----
    hipMemsetAsync(nsum, 0, FEAT_BYTES, stream);
    scatter_kernel<<<edge_blocks, 256, 0, stream>>>(h1, src, dst, nsum, nullptr, N_EDGES);
    norm_mean_kernel<<<elem_blocks, 256, 0, stream>>>(nsum, deg, hnorm); // hnorm = mean
    gemm2_kernel<<<gemm_blocks, 256, 0, stream>>>(h1, hnorm, W2s, W2n, b2, h2);

    // ---- edge scoring ----
    score_kernel<<<edge_blocks, 256, 0, stream>>>(h2, Wpu, Wpv, bp, src, dst, out, N_EDGES);
}